// GaussianBlur2D_19292993093633
// MI455X (gfx1250) — compile-verified
//
#include <hip/hip_runtime.h>
#include <hip/hip_bf16.h>

typedef float v2f __attribute__((ext_vector_type(2)));
typedef float v8f __attribute__((ext_vector_type(8)));

#define KSIZE 201
#define KPAD  100
#define NSLAB 54                  // 54*4 = 216 rows = 201 taps + 15 halo
#define S2INV (1.0f / 1250.0f)    // 1/(2*sigma^2), sigma = 25
#define KNORM 0.015957691216057308f  // 1/(sigma*sqrt(2*pi))

// ---------------------------------------------------------------------------
// Pass 1: vertical 201-tap conv. One wave = two adjacent 16x16 tiles.
// D = sum_s A_s x B_s ; A_s = Toeplitz kernel slice with BOTH the tap gate
// and the H zero-pad gate baked in (block-uniform), precomputed in LDS.
// B rows are loaded through a clamped per-block row-offset table, so the
// inner loop has no branches and no masks at all.
// ---------------------------------------------------------------------------
__global__ __launch_bounds__(192) void gauss_vpass(
    const float* __restrict__ x, float* __restrict__ y, int H, int W)
{
    __shared__ float atab[NSLAB * 64];   // [s][lane][v]  13.8 KB
    __shared__ int   roff[NSLAB * 4];    // clamped row element offsets

    const int h0 = blockIdx.y * 16;

    for (int idx = threadIdx.x; idx < NSLAB * 64; idx += 192) {
        int s = idx >> 6, r = idx & 63;
        int lane = r >> 1, v = r & 1;
        int k = 2 * (lane >> 4) + v;        // K index held by this lane/vgpr
        int m = lane & 15;                  // A-matrix row
        int tap = 4 * s + k - m;
        int row = h0 - KPAD + 4 * s + k;
        float val = 0.0f;
        if ((unsigned)tap <= 200u && (unsigned)row < (unsigned)H) {
            float rr = (float)(tap - KPAD);
            val = expf(-rr * rr * S2INV) * KNORM;
        }
        atab[idx] = val;
    }
    for (int j = threadIdx.x; j < NSLAB * 4; j += 192) {
        int row = h0 - KPAD + j;            // row = h0-100+4s+k, j = 4s+k
        row = row < 0 ? 0 : (row >= H ? H - 1 : row);
        roff[j] = row * W;
    }
    __syncthreads();

    const int lane = threadIdx.x & 31;
    const int wave = threadIdx.x >> 5;
    const int half = lane >> 4;
    const int nm   = lane & 15;
    const int w0   = (blockIdx.x * 12 + wave * 2) * 16;   // tiles w0, w0+16

    const float* xt = x + (size_t)blockIdx.z * H * W;
    const int colbase = w0 + nm;

    const v2f*  ap = (const v2f*)atab;     // ap[s*32 + lane]
    const int2* rp = (const int2*)roff;    // rp[s*2 + half]

    v8f c0 = {}, c1 = {};
#pragma unroll 2
    for (int s = 0; s < NSLAB; ++s) {
        v2f  a  = ap[s * 32 + lane];        // ds_load_b64
        int2 ro = rp[s * 2 + half];         // ds_load_b64 (clamped rows)
        int i0 = ro.x + colbase;
        int i1 = ro.y + colbase;
        v2f b0, b1;
        b0.x = xt[i0];       b0.y = xt[i1];
        b1.x = xt[i0 + 16];  b1.y = xt[i1 + 16];
        c0 = __builtin_amdgcn_wmma_f32_16x16x4_f32(false, a, false, b0,
                                                   (short)0, c0, false, false);
        c1 = __builtin_amdgcn_wmma_f32_16x16x4_f32(false, a, false, b1,
                                                   (short)0, c1, false, false);
    }

    float* yt = y + (size_t)blockIdx.z * H * W;
#pragma unroll
    for (int r = 0; r < 8; ++r) {
        const int m = r + 8 * half;         // C/D row layout per ISA 7.12.2
        yt[(size_t)(h0 + m) * W + colbase]      = c0[r];
        yt[(size_t)(h0 + m) * W + colbase + 16] = c1[r];
    }
}

// ---------------------------------------------------------------------------
// Pass 2: horizontal 201-tap conv. Mirror: input pairs in A (single aligned
// b64 load per tile, med3-clamped column, pairwise validity cndmask), and the
// tap-gated Toeplitz kernel in B, precomputed in LDS (block-uniform).
// ---------------------------------------------------------------------------
__global__ __launch_bounds__(192) void gauss_hpass(
    const float* __restrict__ x, float* __restrict__ y, int H, int W)
{
    __shared__ float btab[NSLAB * 64];   // [s][lane][v]

    for (int idx = threadIdx.x; idx < NSLAB * 64; idx += 192) {
        int s = idx >> 6, r = idx & 63;
        int lane = r >> 1, v = r & 1;
        int k = 2 * (lane >> 4) + v;
        int n = lane & 15;                  // B-matrix column
        int tap = 4 * s + k - n;
        float val = 0.0f;
        if ((unsigned)tap <= 200u) {
            float rr = (float)(tap - KPAD);
            val = expf(-rr * rr * S2INV) * KNORM;
        }
        btab[idx] = val;
    }
    __syncthreads();

    const int lane = threadIdx.x & 31;
    const int wave = threadIdx.x >> 5;
    const int half = lane >> 4;
    const int nm   = lane & 15;
    const int h0   = blockIdx.y * 16;
    const int w0   = (blockIdx.x * 12 + wave * 2) * 16;

    const float* xt   = x + (size_t)blockIdx.z * H * W;
    const float* rowp = xt + (size_t)(h0 + nm) * W;   // this lane's input row

    const v2f* bp = (const v2f*)btab;

    v8f c0 = {}, c1 = {};
    int col = w0 - KPAD + 2 * half;        // always even -> b64 aligned
#pragma unroll 2
    for (int s = 0; s < NSLAB; ++s) {
        v2f b = bp[s * 32 + lane];          // ds_load_b64

        // tile 0: columns col, col+1 (pairwise valid since col is even, W even)
        int cc0 = col < 0 ? 0 : (col > W - 2 ? W - 2 : col);   // v_med3_i32
        bool v0 = (unsigned)col < (unsigned)(W - 1);
        v2f t0 = *(const v2f*)(rowp + cc0);                    // global_load_b64
        v2f a0; a0.x = v0 ? t0.x : 0.0f; a0.y = v0 ? t0.y : 0.0f;

        // tile 1: columns col+16, col+17
        int col1 = col + 16;
        int cc1 = col1 < 0 ? 0 : (col1 > W - 2 ? W - 2 : col1);
        bool v1 = (unsigned)col1 < (unsigned)(W - 1);
        v2f t1 = *(const v2f*)(rowp + cc1);
        v2f a1; a1.x = v1 ? t1.x : 0.0f; a1.y = v1 ? t1.y : 0.0f;

        c0 = __builtin_amdgcn_wmma_f32_16x16x4_f32(false, a0, false, b,
                                                   (short)0, c0, false, false);
        c1 = __builtin_amdgcn_wmma_f32_16x16x4_f32(false, a1, false, b,
                                                   (short)0, c1, false, false);
        col += 4;
    }

    float* yt = y + (size_t)blockIdx.z * H * W;
#pragma unroll
    for (int r = 0; r < 8; ++r) {
        const int m = r + 8 * half;
        yt[(size_t)(h0 + m) * W + w0 + nm]      = c0[r];
        yt[(size_t)(h0 + m) * W + w0 + nm + 16] = c1[r];
    }
}

extern "C" void kernel_launch(void* const* d_in, const int* in_sizes, int n_in,
                              void* d_out, int out_size, void* d_ws, size_t ws_size,
                              hipStream_t stream)
{
    (void)in_sizes; (void)n_in; (void)out_size; (void)ws_size;
    const int T = 64, H = 544, W = 960;

    const float* x   = (const float*)d_in[0];
    float*       out = (float*)d_out;
    float*       tmp = (float*)d_ws;   // 133.7 MB intermediate; lives in the
                                       // 192 MB L2 between the two passes

    dim3 block(192, 1, 1);             // 6 waves x 2 tiles = 12 w-tiles/block
    dim3 grid(60 / 12, H / 16, T);     // 5 x 34 x 64, exact fit

    gauss_vpass<<<grid, block, 0, stream>>>(x,   tmp, H, W);
    gauss_hpass<<<grid, block, 0, stream>>>(tmp, out, H, W);
}